// RelationshipSummarizer_798863917140
// MI455X (gfx1250) — compile-verified
//
#include <hip/hip_runtime.h>

#define DD 128
#define KP2 130   // padded LDS row stride (dwords) for K=128 GEMMs
#define KP1 258   // padded LDS row stride (dwords) for K=256 GEMM

typedef __attribute__((ext_vector_type(2))) float v2f;
typedef __attribute__((ext_vector_type(8))) float v8f;

static __device__ __forceinline__ v8f wmma4(v2f a, v2f b, v8f c) {
  // D = A(16x4,f32) * B(4x16,f32) + C(16x16,f32)  -> v_wmma_f32_16x16x4_f32
  return __builtin_amdgcn_wmma_f32_16x16x4_f32(false, a, false, b, (short)0, c, false, false);
}

// ---------------------------------------------------------------- zero
__global__ __launch_bounds__(256) void zero_f4(float4* __restrict__ p, long n4) {
  long i = (long)blockIdx.x * blockDim.x + threadIdx.x;
  long stride = (long)gridDim.x * blockDim.x;
  float4 z = make_float4(0.f, 0.f, 0.f, 0.f);
  for (; i < n4; i += stride) p[i] = z;
}

// ------------------------------------------------------- scatter mean
// one wave per edge: lane handles 4 contiguous floats (float4 gather,
// 4x global_atomic_add_f32 scatter); lane0 bumps the count.
__global__ __launch_bounds__(256) void scatter_mean(
    const float* __restrict__ x, const int* __restrict__ ei, int E,
    float* __restrict__ agg, float* __restrict__ cnt) {
  int g = blockIdx.x * 256 + threadIdx.x;
  int e = g >> 5, lane = g & 31;
  if (e >= E) return;
  int src = ei[e];
  int dst = ei[E + e];
  const float4 v = *(const float4*)(x + (size_t)src * DD + lane * 4);
  float* dp = agg + (size_t)dst * DD + lane * 4;
  atomicAdd(dp + 0, v.x);
  atomicAdd(dp + 1, v.y);
  atomicAdd(dp + 2, v.z);
  atomicAdd(dp + 3, v.w);
  if (lane == 0) atomicAdd(cnt + dst, 1.0f);
}

// --------------------------------------------- GEMM1: rel = [agg/cnt | x] @ [Wl;Wr] + bl
__global__ __launch_bounds__(256) void gemm_rel(
    const float* __restrict__ agg, const float* __restrict__ cnt,
    const float* __restrict__ x, const float* __restrict__ Wl,
    const float* __restrict__ Wr, const float* __restrict__ bl,
    float* __restrict__ rel, int mtiles) {
  extern __shared__ float bt[];  // B^T: [128 cols][KP1], B = [Wl;Wr] (256x128)
  for (int idx = threadIdx.x; idx < 256 * DD; idx += 256) {
    int k = idx >> 7, n = idx & 127;
    bt[n * KP1 + k] = (k < DD) ? Wl[k * DD + n] : Wr[(k - DD) * DD + n];
  }
  __syncthreads();
  int wave = threadIdx.x >> 5, lane = threadIdx.x & 31;
  int tile = blockIdx.x * 8 + wave;
  if (tile >= mtiles) return;
  size_t row0 = (size_t)tile * 16;
  int mr = lane & 15, kh = lane >> 4, klo = kh * 2;
  size_t m = row0 + mr;
  float inv = 1.0f / fmaxf(cnt[m], 1.0f);
  const float* arow = agg + m * DD;
  const float* xrow = x + m * DD;
  v8f acc[8];
#pragma unroll
  for (int t = 0; t < 8; ++t) acc[t] = (v8f){0.f, 0.f, 0.f, 0.f, 0.f, 0.f, 0.f, 0.f};

#pragma unroll 4
  for (int kk = 0; kk < DD; kk += 4) {   // K = 0..127 : mean-aggregated part
    v2f a = *(const v2f*)(arow + kk + klo);
    a = a * inv;
    const float* bp = bt + mr * KP1 + kk + klo;
#pragma unroll
    for (int nt = 0; nt < 8; ++nt)
      acc[nt] = wmma4(a, *(const v2f*)(bp + nt * 16 * KP1), acc[nt]);
  }
#pragma unroll 4
  for (int kk = 0; kk < DD; kk += 4) {   // K = 128..255 : self part
    v2f a = *(const v2f*)(xrow + kk + klo);
    const float* bp = bt + mr * KP1 + DD + kk + klo;
#pragma unroll
    for (int nt = 0; nt < 8; ++nt)
      acc[nt] = wmma4(a, *(const v2f*)(bp + nt * 16 * KP1), acc[nt]);
  }
#pragma unroll
  for (int nt = 0; nt < 8; ++nt) {
    int n = nt * 16 + mr;
    float bb = bl[n];
#pragma unroll
    for (int v = 0; v < 8; ++v)
      rel[(row0 + v + 8 * kh) * DD + n] = acc[nt][v] + bb;
  }
}

// --------------------------------------------- GEMM2: hpre += rel @ P1_r
__global__ __launch_bounds__(256) void gemm_acc(
    const float* __restrict__ A, const float* __restrict__ B,
    float* __restrict__ Cio, int mtiles) {
  extern __shared__ float bt[];  // B^T: [128][KP2]
  for (int idx = threadIdx.x; idx < DD * DD; idx += 256) {
    int k = idx >> 7, n = idx & 127;
    bt[n * KP2 + k] = B[k * DD + n];
  }
  __syncthreads();
  int wave = threadIdx.x >> 5, lane = threadIdx.x & 31;
  int tile = blockIdx.x * 8 + wave;
  if (tile >= mtiles) return;
  size_t row0 = (size_t)tile * 16;
  int mr = lane & 15, kh = lane >> 4, klo = kh * 2;
  const float* arow = A + (row0 + mr) * DD;
  v8f acc[8];
#pragma unroll
  for (int t = 0; t < 8; ++t) acc[t] = (v8f){0.f, 0.f, 0.f, 0.f, 0.f, 0.f, 0.f, 0.f};

#pragma unroll 4
  for (int kk = 0; kk < DD; kk += 4) {
    v2f a = *(const v2f*)(arow + kk + klo);
    const float* bp = bt + mr * KP2 + kk + klo;
#pragma unroll
    for (int nt = 0; nt < 8; ++nt)
      acc[nt] = wmma4(a, *(const v2f*)(bp + nt * 16 * KP2), acc[nt]);
  }
#pragma unroll
  for (int nt = 0; nt < 8; ++nt) {
    int n = nt * 16 + mr;
#pragma unroll
    for (int v = 0; v < 8; ++v) {
      size_t o = (row0 + v + 8 * kh) * DD + n;
      Cio[o] = Cio[o] + acc[nt][v];
    }
  }
}

// --------------------------------------------- GEMM3: out = relu(hpre + b1) @ P2 + b2
__global__ __launch_bounds__(256) void gemm_out(
    const float* __restrict__ hpre, const float* __restrict__ b1,
    const float* __restrict__ P2, const float* __restrict__ b2,
    float* __restrict__ out, int mtiles) {
  extern __shared__ float bt[];  // P2^T: [128][KP2]
  for (int idx = threadIdx.x; idx < DD * DD; idx += 256) {
    int k = idx >> 7, n = idx & 127;
    bt[n * KP2 + k] = P2[k * DD + n];
  }
  __syncthreads();
  int wave = threadIdx.x >> 5, lane = threadIdx.x & 31;
  int tile = blockIdx.x * 8 + wave;
  if (tile >= mtiles) return;
  size_t row0 = (size_t)tile * 16;
  int mr = lane & 15, kh = lane >> 4, klo = kh * 2;
  const float* hrow = hpre + (row0 + mr) * DD;
  v8f acc[8];
#pragma unroll
  for (int t = 0; t < 8; ++t) acc[t] = (v8f){0.f, 0.f, 0.f, 0.f, 0.f, 0.f, 0.f, 0.f};

#pragma unroll 4
  for (int kk = 0; kk < DD; kk += 4) {
    v2f h = *(const v2f*)(hrow + kk + klo);
    v2f bb = *(const v2f*)(b1 + kk + klo);
    v2f a;
    a.x = fmaxf(h.x + bb.x, 0.0f);   // fused ReLU(h + b1)
    a.y = fmaxf(h.y + bb.y, 0.0f);
    const float* bp = bt + mr * KP2 + kk + klo;
#pragma unroll
    for (int nt = 0; nt < 8; ++nt)
      acc[nt] = wmma4(a, *(const v2f*)(bp + nt * 16 * KP2), acc[nt]);
  }
#pragma unroll
  for (int nt = 0; nt < 8; ++nt) {
    int n = nt * 16 + mr;
    float bb = b2[n];
#pragma unroll
    for (int v = 0; v < 8; ++v)
      out[(row0 + v + 8 * kh) * DD + n] = acc[nt][v] + bb;
  }
}

// ------------------------------------------------------------ launch
extern "C" void kernel_launch(void* const* d_in, const int* in_sizes, int n_in,
                              void* d_out, int out_size, void* d_ws, size_t ws_size,
                              hipStream_t stream) {
  const float* x   = (const float*)d_in[0];
  const int*   e0  = (const int*)d_in[1];
  const int*   e1  = (const int*)d_in[2];
  const int*   e2  = (const int*)d_in[3];
  const float* W_l = (const float*)d_in[4];
  const float* b_l = (const float*)d_in[5];
  const float* W_r = (const float*)d_in[6];
  const float* P1  = (const float*)d_in[7];
  const float* b1  = (const float*)d_in[8];
  const float* P2  = (const float*)d_in[9];
  const float* b2  = (const float*)d_in[10];
  float* out = (float*)d_out;

  const int Nn = in_sizes[0] / DD;   // 100000 (divisible by 16)
  const int E  = in_sizes[1] / 2;    // 500000
  const int mtiles = Nn / 16;
  const int gblocks = (mtiles + 7) / 8;
  const size_t nd = (size_t)Nn * DD;

  float* agg  = (float*)d_ws;        // [N,128]
  float* cnt  = agg + nd;            // [N]
  float* rel  = cnt + Nn;            // [N,128]
  float* hpre = rel + nd;            // [N,128]

  const int* edges[3] = {e0, e1, e2};
  const size_t lds1 = (size_t)KP1 * DD * 4;  // 132096 B
  const size_t lds2 = (size_t)KP2 * DD * 4;  //  66560 B

  zero_f4<<<2048, 256, 0, stream>>>((float4*)hpre, (long)(nd / 4));
  for (int r = 0; r < 3; ++r) {
    zero_f4<<<2048, 256, 0, stream>>>((float4*)agg, (long)((nd + Nn) / 4));
    scatter_mean<<<(E * 32 + 255) / 256, 256, 0, stream>>>(x, edges[r], E, agg, cnt);
    gemm_rel<<<gblocks, 256, lds1, stream>>>(agg, cnt, x,
        W_l + (size_t)r * DD * DD, W_r + (size_t)r * DD * DD, b_l + (size_t)r * DD,
        rel, mtiles);
    gemm_acc<<<gblocks, 256, lds2, stream>>>(rel, P1 + (size_t)r * DD * DD, hpre, mtiles);
  }
  gemm_out<<<gblocks, 256, lds2, stream>>>(hpre, b1, P2, b2, out, mtiles);
}